// HE_FM_74337293959546
// MI455X (gfx1250) — compile-verified
//
#include <hip/hip_runtime.h>

typedef float v2f __attribute__((ext_vector_type(2)));
typedef float v8f __attribute__((ext_vector_type(8)));

#define FM_USER_NUM 1000000
#define FM_EMBED_DIM 64
#define FM_WAVES_PER_BLOCK 8
#define FM_SAMPLES_PER_WAVE 16
#define FM_SAMPLES_PER_BLOCK (FM_WAVES_PER_BLOCK * FM_SAMPLES_PER_WAVE) // 128

// One wave32 handles 16 samples. For each K-chunk of 4:
//   A (16x4 f32): row m = sample m's Vu chunk.  Lane layout (ISA 7.12.2):
//     lanes 0-15: M=lane, VGPR0=K0, VGPR1=K1; lanes 16-31: M=lane-16, VGPR0=K2, VGPR1=K3
//   B (4x16 f32): col n = sample n's Vi chunk.  Mirrored layout:
//     lanes 0-15: N=lane, VGPR0=K0, VGPR1=K1; lanes 16-31: N=lane-16, VGPR0=K2, VGPR1=K3
// => both operands load with identical addressing: float2 at (row + k0 + 2*half).
// Diagonal D[s][s] = dot(Vu[s], Vi[s]) accumulated in fp32 (exact-precision path).
__global__ __launch_bounds__(256) void HE_FM_wmma_kernel(
    const int* __restrict__ input,   // [BATCH,2] int32
    const float* __restrict__ W,     // [USER_NUM+ITEM_NUM, 1]
    const float* __restrict__ bias,  // [1]
    const float* __restrict__ V,     // [USER_NUM+ITEM_NUM, 64]
    float* __restrict__ out)         // [BATCH] (viewed as [BATCH,1])
{
    __shared__ float diag_lds[FM_WAVES_PER_BLOCK][32][8];

    const int tid  = threadIdx.x;
    const int wave = tid >> 5;
    const int lane = tid & 31;
    const int s    = lane & 15;   // sample-in-wave (row for A, col for B)
    const int half = lane >> 4;   // 0: K pair {0,1}; 1: K pair {2,3}

    const int g = blockIdx.x * FM_SAMPLES_PER_BLOCK + wave * FM_SAMPLES_PER_WAVE + s;

    const int uid = input[2 * g];
    const int iid = input[2 * g + 1] + FM_USER_NUM;

    const float* __restrict__ rowU = V + (size_t)uid * FM_EMBED_DIM + 2 * half;
    const float* __restrict__ rowI = V + (size_t)iid * FM_EMBED_DIM + 2 * half;

    v8f c = {};
#pragma unroll
    for (int k0 = 0; k0 < FM_EMBED_DIM; k0 += 4) {
        v2f a = *(const v2f*)(rowU + k0);   // global_load_b64, 8B aligned
        v2f b = *(const v2f*)(rowI + k0);
        // 8 args: (neg_a, A, neg_b, B, c_mod, C, reuse_a, reuse_b)
        c = __builtin_amdgcn_wmma_f32_16x16x4_f32(
            /*neg_a=*/false, a, /*neg_b=*/false, b,
            /*c_mod=*/(short)0, c, /*reuse_a=*/false, /*reuse_b=*/false);
    }

    // Spill the 16x16 accumulator tile so each output lane can grab its diagonal
    // element: D[s][s] lives at VGPR (s&7), lane (s<8 ? s : s+16).
#pragma unroll
    for (int r = 0; r < 8; ++r) diag_lds[wave][lane][r] = c[r];

    __syncthreads();

    if (half == 0) {
        const int srcLane = (s < 8) ? s : (s + 16);
        const int reg     = s & 7;
        const float dot   = diag_lds[wave][srcLane][reg];
        out[g] = W[uid] + W[iid] + bias[0] + dot;
    }
}

extern "C" void kernel_launch(void* const* d_in, const int* in_sizes, int n_in,
                              void* d_out, int out_size, void* d_ws, size_t ws_size,
                              hipStream_t stream) {
    const int*   input = (const int*)d_in[0];   // INPUT [BATCH,2]
    const float* W     = (const float*)d_in[1]; // W     [1.5M,1]
    const float* bias  = (const float*)d_in[2]; // b     [1]
    const float* V     = (const float*)d_in[3]; // V     [1.5M,64]
    float*       out   = (float*)d_out;

    const int batch  = in_sizes[0] / 2;                 // 16384
    const int blocks = batch / FM_SAMPLES_PER_BLOCK;    // 128 (16384 % 128 == 0)

    HE_FM_wmma_kernel<<<blocks, 256, 0, stream>>>(input, W, bias, V, out);
}